// ROIAlign_58231166599559
// MI455X (gfx1250) — compile-verified
//
#include <hip/hip_runtime.h>
#include <hip/hip_bf16.h>

// ROI-Align on gfx1250 via V_WMMA_F32_16X16X4_F32.
//
// Separable formulation: per ROI, 28 y-taps and 28 x-taps (14 sample coords x
// {floor, floor+1}), each tap weight pre-multiplied by validity and the 0.5
// per-axis averaging factor. Then
//   out[ph,pw,c] = sum_{i in 4ph..4ph+3} sum_{j in 4pw..4pw+3}
//                  wy[i] * wx[j] * feat[b, c, y[i], x[j]]
// which maps onto chained f32 WMMAs: M=pw (rows 0..6 live, A rows 7..15 are
// zero), N=16 channels, K=4 x-taps per chunk; accumulate 4 y-taps x 7 chunks
// = 28 WMMAs per (roi, 16-channel tile, ph).

typedef __attribute__((ext_vector_type(2))) float v2f;
typedef __attribute__((ext_vector_type(8))) float v8f;

#define POOL 7
#define NTAP 28          // POOL * SAMPLING(2) * 2 bilinear taps

__global__ __launch_bounds__(256)
void roi_align_wmma_kernel(const float* __restrict__ feat,
                           const float* __restrict__ rois,
                           const int*   __restrict__ roib,
                           float* __restrict__ out,
                           int C, int H, int W)
{
    const int roi   = blockIdx.x;
    const int chalf = blockIdx.y;          // 0/1 -> channels [0,128) / [128,256)
    const int tid   = threadIdx.x;
    const int lane  = tid & 31;            // wave32
    const int wv    = tid >> 5;            // 0..7 waves per block
    const int laneM = lane & 15;           // M row (A) / N column (B,C,D)
    const int hiHalf = lane >> 4;          // 0: K rows {0,1}; 1: K rows {2,3}

    __shared__ int   sxi[NTAP];
    __shared__ float sxw[NTAP];
    __shared__ int   syi[NTAP];
    __shared__ float syw[NTAP];

    // ---- per-ROI tap tables (index + fused weight), one thread per tap ----
    if (tid < NTAP) {
        const int t   = tid;
        const int s   = t >> 1;            // sample index 0..13
        const int par = t & 1;             // 0 = floor tap, 1 = floor+1 tap
        const int p   = s >> 1;            // pool bin 0..6
        const int q   = s & 1;             // sub-sample 0..1
        const float off = (float)p + ((float)q + 0.5f) * 0.5f;

        // x axis
        {
            const float x1 = rois[roi * 4 + 0] * 0.25f;
            const float x2 = rois[roi * 4 + 2] * 0.25f;
            const float bw = fmaxf(x2 - x1, 1.0f) * (1.0f / (float)POOL);
            const float xs = x1 + off * bw;
            const float vx = (xs >= -1.0f && xs <= (float)W) ? 0.5f : 0.0f;
            const float xc = fminf(fmaxf(xs, 0.0f), (float)(W - 1));
            const int xlo  = (int)floorf(xc);
            const int xhi  = min(xlo + 1, W - 1);
            const float l  = xc - (float)xlo;
            sxi[t] = par ? xhi : xlo;
            sxw[t] = (par ? l : (1.0f - l)) * vx;
        }
        // y axis
        {
            const float y1 = rois[roi * 4 + 1] * 0.25f;
            const float y2 = rois[roi * 4 + 3] * 0.25f;
            const float bh = fmaxf(y2 - y1, 1.0f) * (1.0f / (float)POOL);
            const float ys = y1 + off * bh;
            const float vy = (ys >= -1.0f && ys <= (float)H) ? 0.5f : 0.0f;
            const float yc = fminf(fmaxf(ys, 0.0f), (float)(H - 1));
            const int ylo  = (int)floorf(yc);
            const int yhi  = min(ylo + 1, H - 1);
            const float l  = yc - (float)ylo;
            syi[t] = par ? yhi : ylo;
            syw[t] = (par ? l : (1.0f - l)) * vy;
        }
    }
    __syncthreads();

    const int b = roib[roi];
    const int c = chalf * 128 + wv * 16 + laneM;   // this lane's channel
    const size_t HW = (size_t)H * (size_t)W;
    const float* chan  = feat + ((size_t)b * (size_t)C + (size_t)c) * HW;
    float*       obase = out + ((size_t)roi * (size_t)C + (size_t)c) * (POOL * POOL);

    for (int ph = 0; ph < POOL; ++ph) {
        v8f acc = {0.f, 0.f, 0.f, 0.f, 0.f, 0.f, 0.f, 0.f};

        #pragma unroll
        for (int s4 = 0; s4 < 4; ++s4) {
            const int   iy  = ph * 4 + s4;          // y-tap index for this bin
            const float wy  = syw[iy];
            const float* row = chan + (size_t)syi[iy] * (size_t)W;

            #pragma unroll
            for (int k = 0; k < POOL; ++k) {
                // This lane's two K rows of the chunk:
                //   lanes 0-15  -> K rows {0,1} => taps 4k+0, 4k+1
                //   lanes 16-31 -> K rows {2,3} => taps 4k+2, 4k+3
                const int j   = 4 * k + 2 * hiHalf;
                const int   xiA = sxi[j];
                const int   xiB = sxi[j + 1];
                const float xwA = sxw[j];
                const float xwB = sxw[j + 1];

                // B matrix (4x16, K x channels): gathered features
                v2f Bm;
                Bm.x = row[xiA];
                Bm.y = row[xiB];

                // A matrix (16x4): row m carries weights only when m == k,
                // so D row m accumulates exactly the pw==k contributions.
                const float msk = (laneM == k) ? wy : 0.0f;
                v2f Am;
                Am.x = msk * xwA;
                Am.y = msk * xwB;

                acc = __builtin_amdgcn_wmma_f32_16x16x4_f32(
                        /*neg_a=*/false, Am, /*neg_b=*/false, Bm,
                        /*c_mod=*/(short)0, acc,
                        /*reuse_a=*/false, /*reuse_b=*/false);
            }
        }

        // D layout: VGPR v, lanes 0-15 hold M=v (pw=v), N=laneM (channel).
        // Rows 7..15 are exact zeros (A rows 7..15 were zero) and unused.
        if (lane < 16) {
            float* o = obase + ph * POOL;
            #pragma unroll
            for (int v = 0; v < POOL; ++v) {
                o[v] = acc[v];
            }
        }
        __builtin_amdgcn_s_barrier();  // keep waves loosely in phase for L2/L0 reuse
    }
}

extern "C" void kernel_launch(void* const* d_in, const int* in_sizes, int n_in,
                              void* d_out, int out_size, void* d_ws, size_t ws_size,
                              hipStream_t stream) {
    (void)n_in; (void)out_size; (void)d_ws; (void)ws_size;
    const float* feat = (const float*)d_in[0];   // [4, 256, 200, 304] fp32
    const float* rois = (const float*)d_in[1];   // [N, 4] fp32
    const int*   roib = (const int*)d_in[2];     // [N] int32
    float* out = (float*)d_out;                  // [N, 256, 7, 7] fp32

    const int N = in_sizes[2];                   // 1024
    const int C = 256, H = 200, W = 304;

    dim3 grid((unsigned)N, 2, 1);                // (roi, channel-half)
    dim3 block(256, 1, 1);                       // 8 wave32: 8 x 16 = 128 channels
    roi_align_wmma_kernel<<<grid, block, 0, stream>>>(feat, rois, roib, out, C, H, W);
}